// GraphSAGEModel_56212531970402
// MI455X (gfx1250) — compile-verified
//
#include <hip/hip_runtime.h>
#include <hip/hip_bf16.h>

typedef __attribute__((ext_vector_type(2))) float v2f;
typedef __attribute__((ext_vector_type(8))) float v8f;

#define N_NODES 200000
#define N_EDGES 6400000
#define IN_C 30
#define HID_C 32
#define OUT_C 2

// ---------------------------------------------------------------------------
// Zero a float buffer (grid-stride).
// ---------------------------------------------------------------------------
__global__ void zero_f32(float* __restrict__ p, int n) {
    int i = blockIdx.x * blockDim.x + threadIdx.x;
    int stride = gridDim.x * blockDim.x;
    for (; i < n; i += stride) p[i] = 0.0f;
}

// ---------------------------------------------------------------------------
// Scatter layer 1: agg[dst, 0:30] += x[src, 0:30]; cnt[dst] += 1.
// 8 threads per edge, 4 channels each (channel space padded to 32).
// Gather/atomic targets fit in the 192MB L2.
// ---------------------------------------------------------------------------
__global__ void scatter1(const float* __restrict__ x, const int* __restrict__ ei,
                         float* __restrict__ agg, float* __restrict__ cnt) {
    long long tid = (long long)blockIdx.x * blockDim.x + threadIdx.x;
    long long e = tid >> 3;
    int cg = (int)(tid & 7);
    if (e >= N_EDGES) return;
    int src = ei[e];
    int dst = ei[N_EDGES + e];
    const float* xs = x + (long long)src * IN_C;
    float* ad = agg + (long long)dst * HID_C;   // stride 32
    int c0 = cg * 4;
#pragma unroll
    for (int i = 0; i < 4; ++i) {
        int c = c0 + i;
        if (c < IN_C) atomicAdd(&ad[c], xs[c]);
    }
    if (cg == 0) atomicAdd(&cnt[dst], 1.0f);
}

// ---------------------------------------------------------------------------
// Scatter layer 2: agg[dst, 0:32] += h[src, 0:32].
// ---------------------------------------------------------------------------
__global__ void scatter2(const float* __restrict__ h, const int* __restrict__ ei,
                         float* __restrict__ agg) {
    long long tid = (long long)blockIdx.x * blockDim.x + threadIdx.x;
    long long e = tid >> 3;
    int cg = (int)(tid & 7);
    if (e >= N_EDGES) return;
    int src = ei[e];
    int dst = ei[N_EDGES + e];
    const float* hs = h + (long long)src * HID_C;
    float* ad = agg + (long long)dst * HID_C;
    int c0 = cg * 4;
#pragma unroll
    for (int i = 0; i < 4; ++i) atomicAdd(&ad[c0 + i], hs[c0 + i]);
}

// ---------------------------------------------------------------------------
// Layer 1: h = relu((agg/max(cnt,1)) @ W1l + b1 + x @ W1r)
// One wave per 16-node tile; fp32 WMMA 16x16x4, K padded 30->32.
//   A(m,k): lane = m + 16*(k>=2), vgpr = k%2
//   B(k,n): lane = n + 16*(k>=2), vgpr = k%2
//   C(m,n): lane = n + 16*(m>=8), vgpr = m%8
// ---------------------------------------------------------------------------
__global__ void layer1(const float* __restrict__ x, const float* __restrict__ agg,
                       const float* __restrict__ cnt,
                       const float* __restrict__ W1l, const float* __restrict__ b1,
                       const float* __restrict__ W1r, float* __restrict__ h) {
    int wid  = (blockIdx.x * blockDim.x + threadIdx.x) >> 5;
    int lane = threadIdx.x & 31;
    int node0 = wid * 16;
    if (node0 >= N_NODES) return;              // wave-uniform exit (EXEC stays all-1)
    int n  = lane & 15;                        // column (B/C) / row (A) index
    int kh = lane >> 4;                        // K half selector
    long long row = node0 + n;                 // A-matrix source row for this lane
    float scale = 1.0f / fmaxf(cnt[row], 1.0f);
    const float* arow = agg + row * HID_C;
    const float* xrow = x + row * IN_C;

    v8f acc0, acc1;
#pragma unroll
    for (int v = 0; v < 8; ++v) { acc0[v] = b1[n]; acc1[v] = b1[16 + n]; }

    // (agg * scale) @ W1l   -- agg channels 30,31 are zero-initialized
#pragma unroll
    for (int k0 = 0; k0 < 32; k0 += 4) {
        int ka = k0 + 2 * kh, kb = ka + 1;
        v2f a; a.x = arow[ka] * scale; a.y = arow[kb] * scale;
        v2f b0, b1f;
        b0.x  = (ka < IN_C) ? W1l[ka * HID_C + n]        : 0.0f;
        b0.y  = (kb < IN_C) ? W1l[kb * HID_C + n]        : 0.0f;
        b1f.x = (ka < IN_C) ? W1l[ka * HID_C + 16 + n]   : 0.0f;
        b1f.y = (kb < IN_C) ? W1l[kb * HID_C + 16 + n]   : 0.0f;
        acc0 = __builtin_amdgcn_wmma_f32_16x16x4_f32(false, a, false, b0,  (short)0, acc0, false, false);
        acc1 = __builtin_amdgcn_wmma_f32_16x16x4_f32(false, a, false, b1f, (short)0, acc1, false, false);
    }
    // x @ W1r
#pragma unroll
    for (int k0 = 0; k0 < 32; k0 += 4) {
        int ka = k0 + 2 * kh, kb = ka + 1;
        v2f a;
        a.x = (ka < IN_C) ? xrow[ka] : 0.0f;
        a.y = (kb < IN_C) ? xrow[kb] : 0.0f;
        v2f b0, b1f;
        b0.x  = (ka < IN_C) ? W1r[ka * HID_C + n]      : 0.0f;
        b0.y  = (kb < IN_C) ? W1r[kb * HID_C + n]      : 0.0f;
        b1f.x = (ka < IN_C) ? W1r[ka * HID_C + 16 + n] : 0.0f;
        b1f.y = (kb < IN_C) ? W1r[kb * HID_C + 16 + n] : 0.0f;
        acc0 = __builtin_amdgcn_wmma_f32_16x16x4_f32(false, a, false, b0,  (short)0, acc0, false, false);
        acc1 = __builtin_amdgcn_wmma_f32_16x16x4_f32(false, a, false, b1f, (short)0, acc1, false, false);
    }
    // relu + store
#pragma unroll
    for (int v = 0; v < 8; ++v) {
        long long r = node0 + v + 8 * kh;
        h[r * HID_C + n]      = fmaxf(acc0[v], 0.0f);
        h[r * HID_C + 16 + n] = fmaxf(acc1[v], 0.0f);
    }
}

// ---------------------------------------------------------------------------
// Layer 2: o = (agg/max(cnt,1)) @ W2l + b2 + h @ W2r  (2 cols), then
// log_softmax over the 2 logits via shfl_xor(1) pairing, write d_out.
// ---------------------------------------------------------------------------
__global__ void layer2(const float* __restrict__ h, const float* __restrict__ agg,
                       const float* __restrict__ cnt,
                       const float* __restrict__ W2l, const float* __restrict__ b2,
                       const float* __restrict__ W2r, float* __restrict__ out) {
    int wid  = (blockIdx.x * blockDim.x + threadIdx.x) >> 5;
    int lane = threadIdx.x & 31;
    int node0 = wid * 16;
    if (node0 >= N_NODES) return;              // wave-uniform exit
    int n  = lane & 15;
    int kh = lane >> 4;
    long long row = node0 + n;
    float scale = 1.0f / fmaxf(cnt[row], 1.0f);
    const float* arow = agg + row * HID_C;
    const float* hrow = h + row * HID_C;

    v8f acc;
#pragma unroll
    for (int v = 0; v < 8; ++v) acc[v] = (n < OUT_C) ? b2[n] : 0.0f;

    // (agg * scale) @ W2l   (K = 32 exactly)
#pragma unroll
    for (int k0 = 0; k0 < 32; k0 += 4) {
        int ka = k0 + 2 * kh, kb = ka + 1;
        v2f a; a.x = arow[ka] * scale; a.y = arow[kb] * scale;
        v2f b;
        b.x = (n < OUT_C) ? W2l[ka * OUT_C + n] : 0.0f;
        b.y = (n < OUT_C) ? W2l[kb * OUT_C + n] : 0.0f;
        acc = __builtin_amdgcn_wmma_f32_16x16x4_f32(false, a, false, b, (short)0, acc, false, false);
    }
    // h @ W2r
#pragma unroll
    for (int k0 = 0; k0 < 32; k0 += 4) {
        int ka = k0 + 2 * kh, kb = ka + 1;
        v2f a; a.x = hrow[ka]; a.y = hrow[kb];
        v2f b;
        b.x = (n < OUT_C) ? W2r[ka * OUT_C + n] : 0.0f;
        b.y = (n < OUT_C) ? W2r[kb * OUT_C + n] : 0.0f;
        acc = __builtin_amdgcn_wmma_f32_16x16x4_f32(false, a, false, b, (short)0, acc, false, false);
    }
    // log_softmax over the two logit columns (lanes n=0 <-> n=1 pair up;
    // same pairing holds in the upper half-wave for rows m>=8).
#pragma unroll
    for (int v = 0; v < 8; ++v) {
        float mine  = acc[v];
        float other = __shfl_xor(mine, 1, 32);
        float o0 = (n == 0) ? mine : other;
        float o1 = (n == 0) ? other : mine;
        float mx  = fmaxf(o0, o1);
        float lse = mx + __logf(__expf(o0 - mx) + __expf(o1 - mx));
        if (n < OUT_C) {
            long long r = node0 + v + 8 * kh;
            out[r * OUT_C + n] = mine - lse;
        }
    }
}

// ---------------------------------------------------------------------------
extern "C" void kernel_launch(void* const* d_in, const int* in_sizes, int n_in,
                              void* d_out, int out_size, void* d_ws, size_t ws_size,
                              hipStream_t stream) {
    const float* x   = (const float*)d_in[0];
    const int*   ei  = (const int*)d_in[1];
    const float* W1l = (const float*)d_in[2];
    const float* b1  = (const float*)d_in[3];
    const float* W1r = (const float*)d_in[4];
    const float* W2l = (const float*)d_in[5];
    const float* b2  = (const float*)d_in[6];
    const float* W2r = (const float*)d_in[7];
    float* out = (float*)d_out;

    // Workspace partition: agg[N*32] | cnt[N] | h[N*32]  (~52 MB)
    float* agg = (float*)d_ws;
    float* cnt = agg + (size_t)N_NODES * HID_C;
    float* h   = cnt + (size_t)N_NODES;

    // 1. zero agg + cnt (contiguous)
    zero_f32<<<2048, 256, 0, stream>>>(agg, N_NODES * (HID_C + 1));
    // 2. edge scatter for layer 1 (+ degree counts)
    long long sthreads = (long long)N_EDGES * 8;
    int sblocks = (int)((sthreads + 255) / 256);
    scatter1<<<sblocks, 256, 0, stream>>>(x, ei, agg, cnt);
    // 3. layer 1 dense transform (WMMA) -> h
    int tiles  = (N_NODES + 15) / 16;
    int lblocks = (tiles + 7) / 8;              // 8 waves / 256-thread block
    layer1<<<lblocks, 256, 0, stream>>>(x, agg, cnt, W1l, b1, W1r, h);
    // 4. re-zero agg for layer 2
    zero_f32<<<2048, 256, 0, stream>>>(agg, N_NODES * HID_C);
    // 5. edge scatter for layer 2
    scatter2<<<sblocks, 256, 0, stream>>>(h, ei, agg);
    // 6. layer 2 dense transform (WMMA) + log_softmax -> d_out
    layer2<<<lblocks, 256, 0, stream>>>(h, agg, cnt, W2l, b2, W2r, out);
}